// GraphLayer_33998961115155
// MI455X (gfx1250) — compile-verified
//
#include <hip/hip_runtime.h>
#include <hip/hip_bf16.h>

// Problem constants (fixed by the reference)
#define BATCH 4
#define NPTS  8192
#define CH    32
#define KNN   16
#define LDIM  64
#define ODIM  128
#define NCTILES (NPTS / 16)     // 512 column tiles per batch
#define ROWS_PER_WAVE 32        // two 16x16 accumulators per wave
#define NRBLK (NPTS / ROWS_PER_WAVE)

typedef __attribute__((ext_vector_type(2))) float v2f;
typedef __attribute__((ext_vector_type(8))) float v8f;

// ---------------------------------------------------------------------------
// Kernel 0: per-point squared norms  sq[b*N+n] = sum_c x[b,n,c]^2
// ---------------------------------------------------------------------------
__global__ void sq_kernel(const float* __restrict__ x, float* __restrict__ sq, int total) {
    int i = blockIdx.x * blockDim.x + threadIdx.x;
    if (i >= total) return;
    const float* p = x + (size_t)i * CH;
    float s = 0.0f;
#pragma unroll
    for (int c = 0; c < CH; ++c) s += p[c] * p[c];
    sq[i] = s;
}

// ---------------------------------------------------------------------------
// Kernel 1: fused ranking-key GEMM (WMMA f32) + streaming sorted top-16 +
// gather + max-pool.  One wave32 per (batch, 32-row block): two accumulator
// tiles share every B fragment; one lane owns one row's top-16 (no merge).
//
// Key trick: with A' = [-2*x_i, 1] (K=36, zero-padded) and B' = [x_j, sq_j],
// the WMMA accumulator directly equals  key_ij = sq_j - 2*<x_i,x_j>,
// which ranks identically to the true squared distance (sq_i is row-const).
// ---------------------------------------------------------------------------
__global__ __launch_bounds__(32) void knn_pool_kernel(const float* __restrict__ x,
                                                      const float* __restrict__ sq,
                                                      float* __restrict__ pooled) {
    __shared__ float distT[ROWS_PER_WAVE * 19];   // padded, bank-conflict-free
    __shared__ int   selIdx[ROWS_PER_WAVE * KNN];

    const int lane    = threadIdx.x;              // 0..31 (wave32)
    const int rblk    = blockIdx.x % NRBLK;
    const int b       = blockIdx.x / NRBLK;
    const int rowbase = rblk * ROWS_PER_WAVE;
    const float* xb   = x  + (size_t)b * NPTS * CH;
    const float* sqb  = sq + (size_t)b * NPTS;

    const int m = lane & 15;                      // matrix row/col within tile
    const int h = lane >> 4;                      // lane half selects K pair

    // A fragments, pre-scaled by -2: chunks 0..7 real channels, chunk 8 = [1,0]
    v2f afrag0[8], afrag1[8];
#pragma unroll
    for (int c = 0; c < 8; ++c) {
        const float* p0 = xb + (size_t)(rowbase + m) * CH + 4 * c + 2 * h;
        const float* p1 = xb + (size_t)(rowbase + 16 + m) * CH + 4 * c + 2 * h;
        afrag0[c].x = -2.0f * p0[0];
        afrag0[c].y = -2.0f * p0[1];
        afrag1[c].x = -2.0f * p1[0];
        afrag1[c].y = -2.0f * p1[1];
    }
    v2f afragOne;                                 // K=32 row of ones (bias row)
    afragOne.x = (h == 0) ? 1.0f : 0.0f;          // K=32 (h=0) / K=34 (h=1)
    afragOne.y = 0.0f;                            // K=33 / K=35

    // Per-lane sorted (ascending) top-16: key[15] is the current worst.
    float key[KNN];
    int   idx[KNN];
#pragma unroll
    for (int s = 0; s < KNN; ++s) { key[s] = 3.0e38f; idx[s] = 0; }

    for (int jt = 0; jt < NCTILES; ++jt) {
        const int colbase = jt * 16;

        // --- B fragments shared by both accumulators
        v8f acc0 = {};
        v8f acc1 = {};
#pragma unroll
        for (int c = 0; c < 8; ++c) {
            const float* p = xb + (size_t)(colbase + m) * CH + 4 * c + 2 * h;
            v2f bfrag;
            bfrag.x = p[0];
            bfrag.y = p[1];
            acc0 = __builtin_amdgcn_wmma_f32_16x16x4_f32(
                false, afrag0[c], false, bfrag, (short)0, acc0, false, false);
            acc1 = __builtin_amdgcn_wmma_f32_16x16x4_f32(
                false, afrag1[c], false, bfrag, (short)0, acc1, false, false);
        }
        {   // augmented chunk: B row K=32 carries sq_j
            v2f bfrag;
            bfrag.x = (h == 0) ? sqb[colbase + m] : 0.0f;
            bfrag.y = 0.0f;
            acc0 = __builtin_amdgcn_wmma_f32_16x16x4_f32(
                false, afragOne, false, bfrag, (short)0, acc0, false, false);
            acc1 = __builtin_amdgcn_wmma_f32_16x16x4_f32(
                false, afragOne, false, bfrag, (short)0, acc1, false, false);
        }

        // C layout -> row-major LDS: VGPR r, lane => row (r + 8*h), col m
#pragma unroll
        for (int r = 0; r < 8; ++r) {
            distT[(r + 8 * h) * 19 + m]        = acc0[r];
            distT[(r + 8 * h + 16) * 19 + m]   = acc1[r];
        }
        __syncthreads();

        // --- streaming sorted top-16: one lane per row, 16 candidates/tile
#pragma unroll 1
        for (int c = 0; c < 16; ++c) {
            const float d = distT[lane * 19 + c];
            if (d < key[KNN - 1]) {
                float cv = d;
                int   ci = colbase + c;
#pragma unroll
                for (int s = 0; s < KNN; ++s) {
                    const float ks = key[s];
                    const int   is = idx[s];
                    const bool  sw = cv < ks;
                    key[s] = sw ? cv : ks;
                    idx[s] = sw ? ci : is;
                    cv = sw ? ks : cv;
                    ci = sw ? is : ci;
                }
            }
        }
        __syncthreads();
    }

    // --- publish neighbor indices (already the exact top-16 per row)
#pragma unroll
    for (int s = 0; s < KNN; ++s)
        selIdx[lane * KNN + s] = idx[s];
    __syncthreads();

    // --- gather neighbor features + max pool.  lane == channel (CH==32)
    float* poolb = pooled + ((size_t)b * NPTS + rowbase) * CH;
#pragma unroll 1
    for (int r = 0; r < ROWS_PER_WAVE; ++r) {
        float mx = -3.0e38f;
#pragma unroll
        for (int k = 0; k < KNN; ++k) {
            const int j = selIdx[r * KNN + k];
            const float v = xb[(size_t)j * CH + lane];
            mx = fmaxf(mx, v);
        }
        poolb[r * CH + lane] = mx;
    }
}

// ---------------------------------------------------------------------------
// Kernel 2: MLP  pooled[*,32] @ Wl[32,64] + bl  ->  @ Wc[64,128] + bc -> relu
// One wave32 per 16-row tile of the flat (B*N) dimension.
// ---------------------------------------------------------------------------
__global__ __launch_bounds__(32) void mlp_kernel(const float* __restrict__ pooled,
                                                 const float* __restrict__ Wl,
                                                 const float* __restrict__ bl,
                                                 const float* __restrict__ Wc,
                                                 const float* __restrict__ bc,
                                                 float* __restrict__ out) {
    __shared__ float hT[16 * 68];   // 16 x 64 hidden tile, padded

    const int lane = threadIdx.x;
    const int m = lane & 15;
    const int h = lane >> 4;
    const size_t rowbase = (size_t)blockIdx.x * 16;

    // A fragments of pooled tile (k = 32 -> 8 chunks)
    v2f afrag[8];
#pragma unroll
    for (int c = 0; c < 8; ++c) {
        const float* p = pooled + (rowbase + m) * CH + 4 * c + 2 * h;
        afrag[c].x = p[0];
        afrag[c].y = p[1];
    }

    // GEMM1: [16,32] x [32,64]  (4 n-tiles of 16)
#pragma unroll
    for (int t = 0; t < 4; ++t) {
        v8f acc = {};
#pragma unroll
        for (int c = 0; c < 8; ++c) {
            const float* p = Wl + (size_t)(4 * c + 2 * h) * LDIM + t * 16 + m;
            v2f bfrag;
            bfrag.x = p[0];
            bfrag.y = p[LDIM];
            acc = __builtin_amdgcn_wmma_f32_16x16x4_f32(
                false, afrag[c], false, bfrag, (short)0, acc, false, false);
        }
        const float bias = bl[t * 16 + m];
#pragma unroll
        for (int r = 0; r < 8; ++r)
            hT[(r + 8 * h) * 68 + t * 16 + m] = acc[r] + bias;
    }
    __syncthreads();

    // Re-load hidden tile as A fragments (k = 64 -> 16 chunks)
    v2f a2[16];
#pragma unroll
    for (int c = 0; c < 16; ++c) {
        a2[c].x = hT[m * 68 + 4 * c + 2 * h];
        a2[c].y = hT[m * 68 + 4 * c + 2 * h + 1];
    }

    // GEMM2: [16,64] x [64,128] + bias + relu  (8 n-tiles of 16)
    float* outp = out + rowbase * ODIM;
#pragma unroll
    for (int t = 0; t < 8; ++t) {
        v8f acc = {};
#pragma unroll
        for (int c = 0; c < 16; ++c) {
            const float* p = Wc + (size_t)(4 * c + 2 * h) * ODIM + t * 16 + m;
            v2f bfrag;
            bfrag.x = p[0];
            bfrag.y = p[ODIM];
            acc = __builtin_amdgcn_wmma_f32_16x16x4_f32(
                false, a2[c], false, bfrag, (short)0, acc, false, false);
        }
        const float bias = bc[t * 16 + m];
#pragma unroll
        for (int r = 0; r < 8; ++r) {
            const float v = acc[r] + bias;
            outp[(size_t)(r + 8 * h) * ODIM + t * 16 + m] = fmaxf(v, 0.0f);
        }
    }
}

// ---------------------------------------------------------------------------
extern "C" void kernel_launch(void* const* d_in, const int* in_sizes, int n_in,
                              void* d_out, int out_size, void* d_ws, size_t ws_size,
                              hipStream_t stream) {
    const float* x  = (const float*)d_in[0];   // [B,N,32]
    const float* Wl = (const float*)d_in[1];   // [32,64]
    const float* bl = (const float*)d_in[2];   // [64]
    const float* Wc = (const float*)d_in[3];   // [64,128]
    const float* bc = (const float*)d_in[4];   // [128]
    float* out = (float*)d_out;                // [B,N,128]

    // workspace: pooled [B*N*32] then sq [B*N]
    float* pooled = (float*)d_ws;
    float* sqbuf  = pooled + (size_t)BATCH * NPTS * CH;

    const int total = BATCH * NPTS;
    sq_kernel<<<(total + 255) / 256, 256, 0, stream>>>(x, sqbuf, total);
    knn_pool_kernel<<<BATCH * NRBLK, 32, 0, stream>>>(x, sqbuf, pooled);
    mlp_kernel<<<total / 16, 32, 0, stream>>>(pooled, Wl, bl, Wc, bc, out);
}